// GCN_5609227289259
// MI455X (gfx1250) — compile-verified
//
#include <hip/hip_runtime.h>
#include <hip/hip_bf16.h>

typedef __attribute__((ext_vector_type(2))) float v2f;
typedef __attribute__((ext_vector_type(8))) float v8f;

// ---------------------------------------------------------------------------
// Degree / normalization kernels
// ---------------------------------------------------------------------------
__global__ void deg_init(float* __restrict__ deg, int n) {
    int i = blockIdx.x * blockDim.x + threadIdx.x;
    if (i < n) deg[i] = 1.0f;   // self-loop contributes 1 to every node
}

__global__ void deg_count(const int* __restrict__ dst, float* __restrict__ deg, int e) {
    int i = blockIdx.x * blockDim.x + threadIdx.x;
    if (i < e) atomicAdd(&deg[dst[i]], 1.0f);
}

__global__ void deg_rsqrt(float* __restrict__ deg, int n) {
    int i = blockIdx.x * blockDim.x + threadIdx.x;
    if (i < n) deg[i] = rsqrtf(deg[i]);   // deg >= 1 always (self-loops)
}

// ---------------------------------------------------------------------------
// H = X @ W   (X: [N,64] f32 row-major, W: [64,64] f32 row-major)
// One wave computes a 16x16 output tile via V_WMMA_F32_16X16X4_F32,
// looping K in steps of 4 (16 WMMA ops per tile).
//
// A (16x4 f32) lane layout: lanes 0-15 M=0..15 | VGPR0: K=0 (lo half)/K=2 (hi),
//                                                VGPR1: K=1 (lo)/K=3 (hi)
// B (4x16 f32): mirror with N across lanes.
// C/D (16x16 f32, 8 VGPRs): VGPR r -> M = r + (lane>=16 ? 8 : 0), N = lane%16.
// ---------------------------------------------------------------------------
__global__ __launch_bounds__(128) void gemm64(const float* __restrict__ X,
                                              const float* __restrict__ W,
                                              float* __restrict__ H, int n) {
    const int lane = threadIdx.x & 31;
    const int wave = threadIdx.x >> 5;          // 4 waves -> 4 column tiles
    const int row0 = blockIdx.x * 16;
    const int n0   = wave * 16;
    const int ml   = lane & 15;                 // M (for A) / N (for B) within tile
    const int kofs = (lane >> 4) * 2;           // 0 for lanes 0-15, 2 for lanes 16-31

    int m = row0 + ml;
    if (m >= n) m = n - 1;                      // clamp (N=100000 is 16-divisible anyway)
    const float* __restrict__ xrow = X + (size_t)m * 64;

    v8f c = {};
#pragma unroll
    for (int k0 = 0; k0 < 64; k0 += 4) {
        v2f a, b;
        a.x = xrow[k0 + kofs];
        a.y = xrow[k0 + kofs + 1];
        b.x = W[(size_t)(k0 + kofs) * 64 + n0 + ml];
        b.y = W[(size_t)(k0 + kofs + 1) * 64 + n0 + ml];
        c = __builtin_amdgcn_wmma_f32_16x16x4_f32(false, a, false, b,
                                                  (short)0, c, false, false);
    }

    const int nc    = lane & 15;
    const int mbase = (lane >> 4) * 8;
#pragma unroll
    for (int r = 0; r < 8; ++r) {
        int mm = row0 + mbase + r;
        if (mm < n) H[(size_t)mm * 64 + n0 + nc] = c[r];
    }
}

// ---------------------------------------------------------------------------
// Zero the accumulator
// ---------------------------------------------------------------------------
__global__ void zero_f(float* __restrict__ p, long long n) {
    long long i = (long long)blockIdx.x * blockDim.x + threadIdx.x;
    if (i < n) p[i] = 0.0f;
}

// ---------------------------------------------------------------------------
// Edge scatter: one wave per edge. Lane l handles features 2l, 2l+1.
// 256B coalesced gather of h[src], scaled, scattered with f32 atomics
// (L2 is 192 MB -> the 25.6 MB accumulator stays on-chip).
// ---------------------------------------------------------------------------
__global__ __launch_bounds__(256) void edge_scatter(const float* __restrict__ H,
                                                    const int* __restrict__ src,
                                                    const int* __restrict__ dst,
                                                    const float* __restrict__ dinv,
                                                    float* __restrict__ acc, int e) {
    const int lane = threadIdx.x & 31;
    long long w  = (long long)blockIdx.x * (blockDim.x >> 5) + (threadIdx.x >> 5);
    long long nw = (long long)gridDim.x * (blockDim.x >> 5);
    for (long long i = w; i < e; i += nw) {
        const int s = src[i];
        const int d = dst[i];
        const float nrm = dinv[s] * dinv[d];
        const float2 v = ((const float2*)(H + (size_t)s * 64))[lane];
        float* ap = acc + (size_t)d * 64 + lane * 2;
        atomicAdd(ap,     v.x * nrm);
        atomicAdd(ap + 1, v.y * nrm);
    }
}

// ---------------------------------------------------------------------------
// out = prelu(acc + h * dinv^2 (self-loop) + bias, alpha)
// ---------------------------------------------------------------------------
__global__ void finalize(const float* __restrict__ acc, const float* __restrict__ H,
                         const float* __restrict__ dinv, const float* __restrict__ bias,
                         const float* __restrict__ alpha, float* __restrict__ out,
                         int n) {
    long long t = (long long)blockIdx.x * blockDim.x + threadIdx.x;
    if (t >= (long long)n * 64) return;
    const int i = (int)(t >> 6);
    const int f = (int)(t & 63);
    const float di = dinv[i];
    float v = acc[t] + H[t] * di * di + bias[f];
    const float a = alpha[0];
    out[t] = v >= 0.0f ? v : a * v;
}

// ---------------------------------------------------------------------------
// Driver
// ---------------------------------------------------------------------------
extern "C" void kernel_launch(void* const* d_in, const int* in_sizes, int n_in,
                              void* d_out, int out_size, void* d_ws, size_t ws_size,
                              hipStream_t stream) {
    const float* x  = (const float*)d_in[0];
    const int*   ei = (const int*)d_in[1];
    const float* W1 = (const float*)d_in[2];
    const float* b1 = (const float*)d_in[3];
    const float* a1 = (const float*)d_in[4];
    const float* W2 = (const float*)d_in[5];
    const float* b2 = (const float*)d_in[6];
    const float* a2 = (const float*)d_in[7];

    const int n = in_sizes[0] / 64;
    const int e = in_sizes[1] / 2;
    const int* src = ei;
    const int* dst = ei + e;

    float* out = (float*)d_out;

    // Workspace layout: dinv[n] | H[n*64] | ACC[n*64]
    float* wsf  = (float*)d_ws;
    float* dinv = wsf;
    float* Hbuf = wsf + (((size_t)n + 63) & ~(size_t)63);
    float* Acc  = Hbuf + (size_t)n * 64;

    const int T = 256;
    const int nBlocks  = (n + T - 1) / T;
    const int eBlocks  = (e + T - 1) / T;
    const int nfBlocks = (int)(((long long)n * 64 + T - 1) / T);
    const int gBlocks  = (n + 15) / 16;
    int scBlocks = (int)(((long long)e + 7) / 8);   // 8 waves / block, wave per edge
    if (scBlocks > 65535 * 8) scBlocks = 65535 * 8; // grid-stride covers the rest

    // Degree normalization (shared by both layers)
    deg_init <<<nBlocks, T, 0, stream>>>(dinv, n);
    deg_count<<<eBlocks, T, 0, stream>>>(dst, dinv, e);
    deg_rsqrt<<<nBlocks, T, 0, stream>>>(dinv, n);

    // ---- Layer 1 ----
    gemm64      <<<gBlocks, 128, 0, stream>>>(x, W1, Hbuf, n);
    zero_f      <<<nfBlocks, T, 0, stream>>>(Acc, (long long)n * 64);
    edge_scatter<<<scBlocks, T, 0, stream>>>(Hbuf, src, dst, dinv, Acc, e);
    finalize    <<<nfBlocks, T, 0, stream>>>(Acc, Hbuf, dinv, b1, a1, out, n);

    // ---- Layer 2 (reads layer-1 activations from d_out) ----
    gemm64      <<<gBlocks, 128, 0, stream>>>(out, W2, Hbuf, n);
    zero_f      <<<nfBlocks, T, 0, stream>>>(Acc, (long long)n * 64);
    edge_scatter<<<scBlocks, T, 0, stream>>>(Hbuf, src, dst, dinv, Acc, e);
    finalize    <<<nfBlocks, T, 0, stream>>>(Acc, Hbuf, dinv, b2, a2, out, n);
}